// OctreeConv3D_15479062134906
// MI455X (gfx1250) — compile-verified
//
#include <hip/hip_runtime.h>

typedef float v2f __attribute__((ext_vector_type(2)));
typedef float v8f __attribute__((ext_vector_type(8)));

#define NIN   32
#define NOUT  32
#define TAPS  27
#define GD    128

// ---------------------------------------------------------------------------
// Kernel 1: zero the dense output (harness poisons d_out with 0xAA).
// ---------------------------------------------------------------------------
__global__ void OctreeConv3D_zero_kernel(float4* __restrict__ out, int n4) {
    int i = blockIdx.x * blockDim.x + threadIdx.x;
    int stride = gridDim.x * blockDim.x;
    float4 z = make_float4(0.f, 0.f, 0.f, 0.f);
    for (; i < n4; i += stride) out[i] = z;
}

// ---------------------------------------------------------------------------
// Kernel 2: per-tap GEMM via V_WMMA_F32_16X16X4_F32 + atomic scatter-add.
// One wave handles a tile of 16 voxels. Weights staged in LDS as
// w[tap][nout][nin] so each B fragment is one ds_load_b64.
// ---------------------------------------------------------------------------
__global__ __launch_bounds__(256) void OctreeConv3D_scatter_kernel(
    const float* __restrict__ values,   // [N, NIN]
    const int*   __restrict__ coords,   // [N, 3] (z,y,x)
    const float* __restrict__ kern,     // [3,3,3,NIN,NOUT]
    float*       __restrict__ out,      // [GD^3, NOUT]
    int N, int ntiles)
{
    __shared__ float w[TAPS * NOUT * NIN];   // 110,592 B (<320 KB/WGP on CDNA5)

    // Cooperative transposed load: kern[t, k(nin), n(nout)] -> w[t, n, k]
    for (int idx = threadIdx.x; idx < TAPS * NIN * NOUT; idx += blockDim.x) {
        int t = idx / (NIN * NOUT);
        int r = idx - t * (NIN * NOUT);
        int k = r >> 5;          // nin
        int n = r & 31;          // nout
        w[(t * NOUT + n) * NIN + k] = kern[idx];
    }
    __syncthreads();

    const int lane  = threadIdx.x & 31;
    const int wave  = threadIdx.x >> 5;
    const int gwave = blockIdx.x * (blockDim.x >> 5) + wave;
    const int nwv   = gridDim.x * (blockDim.x >> 5);
    const int koff  = (lane >> 4) << 1;   // 0 for lanes 0-15, 2 for lanes 16-31
    const int nloc  = lane & 15;

    const v2f azero = {0.f, 0.f};

    for (int tile = gwave; tile < ntiles; tile += nwv) {
        int  row   = tile * 16 + nloc;
        bool valid = row < N;
        int  rowc  = valid ? row : (N - 1);

        // A fragments: a[c] = values[row][4c+koff .. 4c+koff+1]
        // (16x4 f32 A layout: lanes 0-15 K={0,1}, lanes 16-31 K={2,3})
        v2f a[8];
        const float* vp = values + (size_t)rowc * NIN + koff;
#pragma unroll
        for (int c = 0; c < 8; ++c) {
            v2f t = *(const v2f*)(vp + 4 * c);
            a[c] = valid ? t : azero;
        }

        int z = coords[rowc * 3 + 0];
        int y = coords[rowc * 3 + 1];
        int x = coords[rowc * 3 + 2];

        for (int tz = 0; tz < 3; ++tz)
        for (int ty = 0; ty < 3; ++ty)
        for (int tx = 0; tx < 3; ++tx) {
            int t = (tz * 3 + ty) * 3 + tx;

            int zz = min(max(z + tz - 1, 0), GD - 1);
            int yy = min(max(y + ty - 1, 0), GD - 1);
            int xx = min(max(x + tx - 1, 0), GD - 1);
            int flat = (zz * GD + yy) * GD + xx;   // lane L<16 holds row L's index

            v8f acc0 = {0.f, 0.f, 0.f, 0.f, 0.f, 0.f, 0.f, 0.f};
            v8f acc1 = {0.f, 0.f, 0.f, 0.f, 0.f, 0.f, 0.f, 0.f};

            const float* wt = w + (size_t)t * NOUT * NIN;
#pragma unroll
            for (int c = 0; c < 8; ++c) {
                // B fragments (4x16): lane -> (K = 4c+koff, 4c+koff+1 ; N = nloc [+16])
                v2f b0 = *(const v2f*)(wt + nloc * NIN + 4 * c + koff);
                v2f b1 = *(const v2f*)(wt + (nloc + 16) * NIN + 4 * c + koff);
                acc0 = __builtin_amdgcn_wmma_f32_16x16x4_f32(
                    false, a[c], false, b0, (short)0, acc0, false, false);
                acc1 = __builtin_amdgcn_wmma_f32_16x16x4_f32(
                    false, a[c], false, b1, (short)0, acc1, false, false);
            }

            // Scatter: D vgpr v holds (M=v, N=lane) for lanes 0-15 and
            // (M=v+8, N=lane-16) for lanes 16-31.
#pragma unroll
            for (int v = 0; v < 8; ++v) {
                int src = (lane < 16) ? v : (v + 8);
                int rf  = __shfl(flat, src, 32);
                float* o = out + (size_t)rf * NOUT + nloc;
                unsafeAtomicAdd(o,      acc0[v]);
                unsafeAtomicAdd(o + 16, acc1[v]);
            }
        }
    }
}

// ---------------------------------------------------------------------------
// Kernel 3: out = (out + mask*bias) * mask
// ---------------------------------------------------------------------------
__global__ void OctreeConv3D_finalize_kernel(
    float4* __restrict__ out4, const float* __restrict__ mask,
    const float4* __restrict__ bias4, int total4)
{
    int i = blockIdx.x * blockDim.x + threadIdx.x;
    int stride = gridDim.x * blockDim.x;
    for (; i < total4; i += stride) {
        int cell = i >> 3;        // 32 channels = 8 float4 per cell
        int cg   = i & 7;
        float  m = mask[cell];
        float4 o = out4[i];
        float4 b = bias4[cg];
        o.x = (o.x + m * b.x) * m;
        o.y = (o.y + m * b.y) * m;
        o.z = (o.z + m * b.z) * m;
        o.w = (o.w + m * b.w) * m;
        out4[i] = o;
    }
}

extern "C" void kernel_launch(void* const* d_in, const int* in_sizes, int n_in,
                              void* d_out, int out_size, void* d_ws, size_t ws_size,
                              hipStream_t stream) {
    const float* values = (const float*)d_in[0];
    const int*   coords = (const int*)d_in[1];
    const float* mask   = (const float*)d_in[2];
    const float* kern   = (const float*)d_in[3];
    const float* bias   = (const float*)d_in[4];
    float* out = (float*)d_out;

    int N      = in_sizes[1] / 3;        // 200000
    int ntiles = (N + 15) / 16;
    int n4     = out_size / 4;           // float4 count

    OctreeConv3D_zero_kernel<<<2048, 256, 0, stream>>>((float4*)out, n4);

    int blocks = (ntiles + 7) / 8;
    if (blocks > 1024) blocks = 1024;
    OctreeConv3D_scatter_kernel<<<blocks, 256, 0, stream>>>(
        values, coords, kern, out, N, ntiles);

    OctreeConv3D_finalize_kernel<<<4096, 256, 0, stream>>>(
        (float4*)out, mask, (const float4*)bias, n4);
}